// Social_Aggregator_1821066134227
// MI455X (gfx1250) — compile-verified
//
#include <hip/hip_runtime.h>

// ---------------------------------------------------------------------------
// GraphRec social aggregator for MI455X (gfx1250, wave32, WMMA).
//  V=100000, D=128, B=4096, K=64
//  Kernel A: h1self[b] = W_feat[nodes[b]] @ att1_w[128:256]   (tiny GEMV)
//  Kernel B: per node b:
//    N   = W_feat[neigh_idx[b]]            [64,128]  (gather -> LDS)
//    H1  = relu(N @ W1a + b1 + h1self[b])  [64,128]  (WMMA f32 16x16x4)
//    s   = relu(H1 @ W2 + b2) @ w3 + b3    [64]      (WMMA + shuffle reduce)
//    att = masked softmax(s)
//    out[b] = sum_k att[k] * N[k]
// ---------------------------------------------------------------------------

typedef float v2f __attribute__((ext_vector_type(2)));
typedef float v8f __attribute__((ext_vector_type(8)));

#define DIM   128
#define KNB   64
#define NB    4096
#define ROWP  (DIM + 4)   // padded LDS row stride (floats) -> conflict-free col reads

// ---------------------------------------------------------------------------
// Kernel A: self-hidden precompute. Block handles 8 nodes, 256 threads.
// ---------------------------------------------------------------------------
__global__ __launch_bounds__(256)
void self_hidden_kernel(const float* __restrict__ W_feat,
                        const float* __restrict__ att1_w,   // [256][128]
                        const int*   __restrict__ nodes,
                        float*       __restrict__ h1self)   // [B][128]
{
    __shared__ float su[8][DIM];
    const int b0 = blockIdx.x * 8;
    const int t  = threadIdx.x;

    // load 8 self-embedding rows (8*128 = 1024 floats, 4 per thread)
    for (int e = t; e < 8 * DIM; e += 256) {
        const int r = e >> 7;         // e / 128
        const int d = e & (DIM - 1);
        su[r][d] = W_feat[(size_t)nodes[b0 + r] * DIM + d];
    }
    __syncthreads();

    const float* __restrict__ W1b = att1_w + DIM * DIM;  // rows 128..255
    const int col  = t & (DIM - 1);
    const int rgrp = t >> 7;          // 0..1 -> nodes rgrp*4 .. rgrp*4+3

    float acc[4] = {0.f, 0.f, 0.f, 0.f};
    for (int e = 0; e < DIM; ++e) {
        const float w = W1b[e * DIM + col];   // coalesced
        #pragma unroll
        for (int r = 0; r < 4; ++r)
            acc[r] = fmaf(su[rgrp * 4 + r][e], w, acc[r]);  // LDS broadcast
    }
    #pragma unroll
    for (int r = 0; r < 4; ++r)
        h1self[(size_t)(b0 + rgrp * 4 + r) * DIM + col] = acc[r];
}

// ---------------------------------------------------------------------------
// Kernel B: attention MLP + softmax + weighted sum. One block per node.
// 256 threads = 8 wave32 waves. Wave w: M-tile mt = w/2 (16 rows),
// N-group ng = w&1 (4 consecutive 16-col tiles).
// ---------------------------------------------------------------------------
__global__ __launch_bounds__(256)
void graphrec_attn_kernel(const float* __restrict__ W_feat,
                          const float* __restrict__ att1_w,   // [256][128]
                          const float* __restrict__ att1_b,   // [128]
                          const float* __restrict__ att2_w,   // [128][128]
                          const float* __restrict__ att2_b,   // [128]
                          const float* __restrict__ att3_w,   // [128]
                          const float* __restrict__ att3_b,   // [1]
                          const int*   __restrict__ neigh_idx,// [B][64]
                          const int*   __restrict__ lengths,  // [B]
                          const float* __restrict__ h1self,   // [B][128]
                          float*       __restrict__ out)      // [B][128]
{
    __shared__ float sNeigh[KNB][ROWP];   // 33 KB  neighbor embeddings
    __shared__ float sW[DIM][DIM];        // 64 KB  W1a, then W2
    __shared__ float sH[KNB][ROWP];       // 33 KB  H1
    __shared__ float sBiasSelf[DIM];      // b1 + h1self[b]
    __shared__ float sB2[DIM];
    __shared__ float sW3[DIM];
    __shared__ float sS[KNB];             // scores
    __shared__ float sAtt[KNB];           // softmax weights
    __shared__ int   sIdx[KNB];

    const int b    = blockIdx.x;
    const int t    = threadIdx.x;
    const int lane = t & 31;
    const int wave = t >> 5;
    const int hi   = lane >> 4;       // half-wave select (K pair / M+8)
    const int ln   = lane & 15;       // 0..15

    if (t < KNB) { sIdx[t] = neigh_idx[b * KNB + t]; sS[t] = 0.0f; }
    if (t < DIM) {
        sBiasSelf[t] = att1_b[t] + h1self[(size_t)b * DIM + t];
        sB2[t]       = att2_b[t];
        sW3[t]       = att3_w[t];
    }
    __syncthreads();

    // ---- gather 64 neighbor rows (float4, coalesced 512B per row) ----
    for (int e = t; e < KNB * (DIM / 4); e += 256) {
        const int k  = e >> 5;        // e / 32
        const int d4 = e & 31;
        const float4 v = ((const float4*)(W_feat + (size_t)sIdx[k] * DIM))[d4];
        *(float4*)&sNeigh[k][d4 * 4] = v;
    }
    // ---- stage W1a (rows 0..127 of att1_w) ----
    for (int e = t; e < DIM * DIM / 4; e += 256) {
        *(float4*)(&sW[0][0] + e * 4) = ((const float4*)att1_w)[e];
    }
    __syncthreads();

    const int mt   = wave >> 1;       // 0..3
    const int ng   = wave & 1;        // 0..1
    const int mrow = mt * 16;

    const v8f vzero = {0.f, 0.f, 0.f, 0.f, 0.f, 0.f, 0.f, 0.f};

    // =================== GEMM1: H1 = relu(N @ W1a + bias) ===================
    {
        v8f acc[4] = {vzero, vzero, vzero, vzero};
        for (int k0 = 0; k0 < DIM; k0 += 4) {
            const int ka = k0 + 2 * hi;
            v2f a;
            a.x = sNeigh[mrow + ln][ka];
            a.y = sNeigh[mrow + ln][ka + 1];
            #pragma unroll
            for (int j = 0; j < 4; ++j) {
                const int n = (ng * 4 + j) * 16 + ln;
                v2f bb;
                bb.x = sW[ka][n];
                bb.y = sW[ka + 1][n];
                acc[j] = __builtin_amdgcn_wmma_f32_16x16x4_f32(
                    false, a, false, bb, (short)0, acc[j], false, false);
            }
        }
        // bias + relu + store H1
        #pragma unroll
        for (int j = 0; j < 4; ++j) {
            const int n  = (ng * 4 + j) * 16 + ln;
            const float bs = sBiasSelf[n];
            #pragma unroll
            for (int r = 0; r < 8; ++r) {
                const int m = mrow + hi * 8 + r;
                const float v = acc[j][r] + bs;
                sH[m][n] = v > 0.f ? v : 0.f;
            }
        }
    }
    __syncthreads();

    // ---- overwrite weight buffer with W2 ----
    for (int e = t; e < DIM * DIM / 4; e += 256) {
        *(float4*)(&sW[0][0] + e * 4) = ((const float4*)att2_w)[e];
    }
    __syncthreads();

    // ========== GEMM2: s = relu(H1 @ W2 + b2) . w3  (fold in registers) ====
    {
        v8f acc[4] = {vzero, vzero, vzero, vzero};
        for (int k0 = 0; k0 < DIM; k0 += 4) {
            const int ka = k0 + 2 * hi;
            v2f a;
            a.x = sH[mrow + ln][ka];
            a.y = sH[mrow + ln][ka + 1];
            #pragma unroll
            for (int j = 0; j < 4; ++j) {
                const int n = (ng * 4 + j) * 16 + ln;
                v2f bb;
                bb.x = sW[ka][n];
                bb.y = sW[ka + 1][n];
                acc[j] = __builtin_amdgcn_wmma_f32_16x16x4_f32(
                    false, a, false, bb, (short)0, acc[j], false, false);
            }
        }
        // per-lane partial score contribution for 8 rows
        float part[8];
        #pragma unroll
        for (int r = 0; r < 8; ++r) part[r] = 0.f;
        #pragma unroll
        for (int j = 0; j < 4; ++j) {
            const int n  = (ng * 4 + j) * 16 + ln;
            const float b2 = sB2[n];
            const float w3 = sW3[n];
            #pragma unroll
            for (int r = 0; r < 8; ++r) {
                float v = acc[j][r] + b2;
                v = v > 0.f ? v : 0.f;
                part[r] = fmaf(v, w3, part[r]);
            }
        }
        // reduce over the 16 lanes that share the same row m
        #pragma unroll
        for (int r = 0; r < 8; ++r) {
            float p = part[r];
            p += __shfl_xor(p, 1);
            p += __shfl_xor(p, 2);
            p += __shfl_xor(p, 4);
            p += __shfl_xor(p, 8);
            if (ln == 0) atomicAdd(&sS[mrow + hi * 8 + r], p);  // ds_add_f32
        }
    }
    __syncthreads();

    // =================== masked softmax over K=64 ===================
    if (t == 0) {
        const int   len = lengths[b];
        const float b3  = att3_b[0];
        float mx = -3.0e38f;
        for (int k = 0; k < KNB; ++k) {
            const float v = (k < len) ? (sS[k] + b3) : -1.0e9f;
            sS[k] = v;
            mx = fmaxf(mx, v);
        }
        float sum = 0.f;
        for (int k = 0; k < KNB; ++k) {
            const float e = __expf(sS[k] - mx);
            sAtt[k] = e;
            sum += e;
        }
        const float inv = 1.0f / sum;
        for (int k = 0; k < KNB; ++k) sAtt[k] *= inv;
    }
    __syncthreads();

    // =================== out[b] = sum_k att[k] * N[k] ===================
    if (t < DIM) {
        float o = 0.f;
        for (int k = 0; k < KNB; ++k)
            o = fmaf(sAtt[k], sNeigh[k][t], o);
        out[(size_t)b * DIM + t] = o;
    }
}

// ---------------------------------------------------------------------------
extern "C" void kernel_launch(void* const* d_in, const int* in_sizes, int n_in,
                              void* d_out, int out_size, void* d_ws, size_t ws_size,
                              hipStream_t stream) {
    (void)in_sizes; (void)n_in; (void)out_size; (void)ws_size;
    const float* W_feat  = (const float*)d_in[0];
    const float* att1_w  = (const float*)d_in[1];
    const float* att1_b  = (const float*)d_in[2];
    const float* att2_w  = (const float*)d_in[3];
    const float* att2_b  = (const float*)d_in[4];
    const float* att3_w  = (const float*)d_in[5];
    const float* att3_b  = (const float*)d_in[6];
    const int*   nodes   = (const int*)d_in[7];
    const int*   neigh   = (const int*)d_in[8];
    const int*   lengths = (const int*)d_in[9];
    float*       out     = (float*)d_out;
    float*       h1self  = (float*)d_ws;   // B*D floats = 2 MB

    self_hidden_kernel<<<NB / 8, 256, 0, stream>>>(W_feat, att1_w, nodes, h1self);
    graphrec_attn_kernel<<<NB, 256, 0, stream>>>(W_feat, att1_w, att1_b,
                                                 att2_w, att2_b, att3_w, att3_b,
                                                 neigh, lengths, h1self, out);
}